// AlphaModel_10952166604948
// MI455X (gfx1250) — compile-verified
//
#include <hip/hip_runtime.h>
#include <math.h>

// AlphaModel edge-update: per-row 3x3 reparameterized matvec + softmax + masked blend.
// Pure HBM-streaming kernel: ~100 B/row traffic vs ~65 flops/row -> bandwidth bound
// (~400 MB/call -> ~17us floor @ 23.3 TB/s). WMMA inapplicable: per-row distinct 3x3
// matrices don't map onto shared-K 16x16 tiles without >10x data inflation.
// gfx1250 paths: TDM tensor_load_to_lds for the shared rel tables, s_wait_tensorcnt,
// NT temporal hints for the one-shot streams (working set > 192MB L2),
// global_prefetch_b8 one grid-stride ahead, wave32 split barriers.

namespace {
constexpr int kP     = 3;
constexpr int kRels  = 20;
constexpr float kZEps   = 0.01f;
constexpr float kCosEps = 0.01f;
constexpr float kScale  = 42.0f;
constexpr int kThreads  = 256;   // 8 wave32s
constexpr int kRowsPerT = 4;     // grid-stride iterations (enables prefetch-ahead)
constexpr int kTabElems = kRels * kP * kP;  // 180 floats = 720 B per table
}

#if defined(__HIP_DEVICE_COMPILE__)
#if __has_builtin(__builtin_amdgcn_tensor_load_to_lds_d2) || \
    __has_builtin(__builtin_amdgcn_tensor_load_to_lds)
#define ALPHA_USE_TDM 1
#endif
#endif

#ifdef ALPHA_USE_TDM
typedef unsigned int u32x4 __attribute__((ext_vector_type(4)));
typedef int          i32x4 __attribute__((ext_vector_type(4)));
typedef int          i32x8 __attribute__((ext_vector_type(8)));

// Byte offset of a __shared__ object within the workgroup's LDS allocation.
__device__ inline unsigned int lds_byte_addr(const void* p) {
    return (unsigned int)(unsigned long long)
        (__attribute__((address_space(3))) const void*)p;
}

// Issue one TDM descriptor: copy `nelem` consecutive f32 from global to LDS.
// D# per CDNA5 ISA ch.8: group0 = {count=1, lds_addr, global_addr[56:0], type=2},
// group1 = {data_size=4B, tensor_dim0=tile_dim0=nelem, tensor_dim1=tile_dim1=1,
//           tensor_dim0_stride=nelem}.  2-D tensor -> groups 2/3 disabled/zero.
__device__ inline void tdm_copy_f32(const float* gsrc, const void* lds_dst, int nelem) {
    const unsigned long long ga = (unsigned long long)gsrc;
    u32x4 g0;
    g0.x = 1u;                                              // count=1 (valid), user mode
    g0.y = lds_byte_addr(lds_dst);                          // lds_addr
    g0.z = (unsigned int)(ga & 0xffffffffu);                // global_addr[31:0]
    g0.w = (unsigned int)((ga >> 32) & 0x01ffffffu)         // global_addr[56:32]
         | (2u << 30);                                      // type=2 ("image")
    i32x8 g1;
    g1[0] = (2 << 16);                                      // data_size = 4 bytes
    g1[1] = (nelem & 0xffff) << 16;                         // tensor_dim0[15:0] @ bit48
    g1[2] = (1 << 16) | ((nelem >> 16) & 0xffff);           // dim0[31:16], tensor_dim1=1
    g1[3] = (nelem & 0xffff) << 16;                         // tile_dim0
    g1[4] = 1;                                              // tile_dim1=1 (tile_dim2=0)
    g1[5] = nelem;                                          // tensor_dim0_stride[31:0]
    g1[6] = 0;
    g1[7] = 0;
#if __has_builtin(__builtin_amdgcn_tensor_load_to_lds_d2)
    __builtin_amdgcn_tensor_load_to_lds_d2(g0, g1, 0);
#elif __has_include(<hip/amd_detail/amd_gfx1250_TDM.h>)
    i32x4 z4 = {0, 0, 0, 0};
    i32x8 z8 = {0, 0, 0, 0, 0, 0, 0, 0};
    __builtin_amdgcn_tensor_load_to_lds(g0, g1, z4, z4, z8, 0);   // clang-23 6-arg form
#else
    i32x4 z4 = {0, 0, 0, 0};
    __builtin_amdgcn_tensor_load_to_lds(g0, g1, z4, z4, 0);       // ROCm 7.2 5-arg form
#endif
}

__device__ inline void tdm_wait_all() {
#if __has_builtin(__builtin_amdgcn_s_wait_tensorcnt)
    __builtin_amdgcn_s_wait_tensorcnt(0);
#else
    asm volatile("s_wait_tensorcnt 0x0" ::: "memory");
#endif
}
#endif  // ALPHA_USE_TDM

__global__ __launch_bounds__(kThreads)
void alpha_update_kernel(const float* __restrict__ prnt,
                         const float* __restrict__ child,
                         const float* __restrict__ rel_mu,
                         const float* __restrict__ rel_sigma,
                         const float* __restrict__ eps_M,
                         const float* __restrict__ beta,
                         const int*   __restrict__ rels,
                         float* __restrict__ o_copy_mask,
                         float* __restrict__ o_cp_copy,
                         float* __restrict__ o_alpha_mask,
                         float* __restrict__ o_alpha,
                         int n)
{
    // Shared, randomly-indexed rel tables -> LDS (2 x 720 B).
    __shared__ float s_mu[kTabElems];
    __shared__ float s_sg[kTabElems];

#ifdef ALPHA_USE_TDM
    // TDM DMA: wave 0 issues both descriptors (TDM ignores EXEC; guard is
    // wave-uniform so other waves branch around), waits TENSORcnt, then the
    // whole block syncs at the workgroup barrier.
    if (threadIdx.x < 32u) {
        tdm_copy_f32(rel_mu,    s_mu, kTabElems);
        tdm_copy_f32(rel_sigma, s_sg, kTabElems);
        tdm_wait_all();
    }
#else
    for (int i = threadIdx.x; i < kTabElems; i += blockDim.x) {
        s_mu[i] = rel_mu[i];
        s_sg[i] = rel_sigma[i];
    }
#endif
    __syncthreads();

    const long long stride = (long long)gridDim.x * blockDim.x;
    for (long long idx = (long long)blockIdx.x * blockDim.x + threadIdx.x;
         idx < (long long)n; idx += stride) {

        // Prefetch next grid-stride iteration's rows (gfx1250 global_prefetch_b8;
        // one byte per lane pulls the whole cacheline).
        const long long nxt = idx + stride;
        if (nxt < (long long)n) {
            __builtin_prefetch(eps_M + nxt * 9, 0, 0);
            __builtin_prefetch(child + nxt * 3, 0, 0);
            __builtin_prefetch(prnt  + nxt * 3, 0, 0);
        }

        // ---- streamed, non-temporal row loads ----
        const float p0 = __builtin_nontemporal_load(prnt + idx * 3 + 0);
        const float p1 = __builtin_nontemporal_load(prnt + idx * 3 + 1);
        const float p2 = __builtin_nontemporal_load(prnt + idx * 3 + 2);
        const float c0 = __builtin_nontemporal_load(child + idx * 3 + 0);
        const float c1 = __builtin_nontemporal_load(child + idx * 3 + 1);
        const float c2 = __builtin_nontemporal_load(child + idx * 3 + 2);
        const float bt = __builtin_nontemporal_load(beta + idx);
        const int   r  = __builtin_nontemporal_load(rels + idx);

        const float* __restrict__ mu = s_mu + r * 9;
        const float* __restrict__ sg = s_sg + r * 9;

        // ---- M = mu + sigma * eps ; logits = M @ child ----
        const float cvec[3] = {c0, c1, c2};
        float lg[3];
        #pragma unroll
        for (int i = 0; i < 3; ++i) {
            float acc = 0.0f;
            #pragma unroll
            for (int j = 0; j < 3; ++j) {
                const float e = __builtin_nontemporal_load(eps_M + idx * 9 + i * 3 + j);
                const float m = fmaf(sg[i * 3 + j], e, mu[i * 3 + j]);
                acc = fmaf(m, cvec[j], acc);
            }
            lg[i] = acc;
        }

        // ---- stable softmax ----
        const float mx  = fmaxf(lg[0], fmaxf(lg[1], lg[2]));
        const float e0  = expf(lg[0] - mx);
        const float e1  = expf(lg[1] - mx);
        const float e2  = expf(lg[2] - mx);
        const float ei  = 1.0f / (e0 + e1 + e2);
        const float cp0 = e0 * ei, cp1 = e1 * ei, cp2 = e2 * ei;

        // ---- masks ----
        const float csum  = c0 + c1 + c2;
        const float psum  = p0 + p1 + p2;
        const bool childm = (csum != 0.0f);
        const bool copym  = childm && (psum == 0.0f);
        const bool alpham = childm && (psum != 0.0f);

        // ---- scale = 42 * cos / H(z) ----
        float z0 = fmaxf(kZEps, p0 + cp0);
        float z1 = fmaxf(kZEps, p1 + cp1);
        float z2 = fmaxf(kZEps, p2 + cp2);
        const float zi = 1.0f / (z0 + z1 + z2);
        z0 *= zi; z1 *= zi; z2 *= zi;
        const float H = -(z0 * logf(z0) + z1 * logf(z1) + z2 * logf(z2));

        const float dot = p0 * cp0 + p1 * cp1 + p2 * cp2;
        const float sqp = p0 * p0 + p1 * p1 + p2 * p2;
        const float sqc = cp0 * cp0 + cp1 * cp1 + cp2 * cp2;
        const float np_ = (sqp == 0.0f) ? 0.0f : sqrtf(sqp);
        const float nc_ = (sqc == 0.0f) ? 0.0f : sqrtf(sqc);
        float nrm = np_ * nc_;
        nrm = (nrm == 0.0f) ? 1.0f : nrm;
        const float cosv = fmaxf(kCosEps, dot / nrm);
        const float s    = kScale * cosv / H;

        // ---- outputs (non-temporal: written once, never re-read on device) ----
        const float ob = 1.0f - bt;
        const float a0 = alpham ? fmaf(ob, p0, bt * cp0) * s : 0.0f;
        const float a1 = alpham ? fmaf(ob, p1, bt * cp1) * s : 0.0f;
        const float a2 = alpham ? fmaf(ob, p2, bt * cp2) * s : 0.0f;

        __builtin_nontemporal_store(copym ? 1.0f : 0.0f, o_copy_mask + idx);
        __builtin_nontemporal_store(alpham ? 1.0f : 0.0f, o_alpha_mask + idx);
        __builtin_nontemporal_store(copym ? cp0 : 0.0f, o_cp_copy + idx * 3 + 0);
        __builtin_nontemporal_store(copym ? cp1 : 0.0f, o_cp_copy + idx * 3 + 1);
        __builtin_nontemporal_store(copym ? cp2 : 0.0f, o_cp_copy + idx * 3 + 2);
        __builtin_nontemporal_store(a0, o_alpha + idx * 3 + 0);
        __builtin_nontemporal_store(a1, o_alpha + idx * 3 + 1);
        __builtin_nontemporal_store(a2, o_alpha + idx * 3 + 2);
    }
}

extern "C" void kernel_launch(void* const* d_in, const int* in_sizes, int n_in,
                              void* d_out, int out_size, void* d_ws, size_t ws_size,
                              hipStream_t stream) {
    (void)n_in; (void)out_size; (void)d_ws; (void)ws_size;
    // setup_inputs() order:
    const float* prnt      = (const float*)d_in[0];   // [N,3]
    const float* child     = (const float*)d_in[1];   // [N,3]
    const float* rel_mu    = (const float*)d_in[2];   // [20,3,3]
    const float* rel_sigma = (const float*)d_in[3];   // [20,3,3]
    const float* eps_M     = (const float*)d_in[4];   // [N,3,3]
    const float* beta      = (const float*)d_in[5];   // [N]
    const int*   rels      = (const int*)d_in[6];     // [N]

    const int n = in_sizes[5];  // N (beta element count)

    float* out          = (float*)d_out;
    float* o_copy_mask  = out;                         // [N]
    float* o_cp_copy    = out + (size_t)n;             // [N,3]
    float* o_alpha_mask = out + (size_t)4 * n;         // [N]
    float* o_alpha      = out + (size_t)5 * n;         // [N,3]

    long long blocks = ((long long)n + (long long)kThreads * kRowsPerT - 1) /
                       ((long long)kThreads * kRowsPerT);
    if (blocks < 1) blocks = 1;
    if (blocks > (1 << 20)) blocks = (1 << 20);

    alpha_update_kernel<<<dim3((unsigned)blocks), dim3(kThreads), 0, stream>>>(
        prnt, child, rel_mu, rel_sigma, eps_M, beta, rels,
        o_copy_mask, o_cp_copy, o_alpha_mask, o_alpha, n);
}